// DynamicDWConv_6253472383673
// MI455X (gfx1250) — compile-verified
//
#include <hip/hip_runtime.h>

// DynamicDWConv for MI455X (gfx1250, wave32).
// b=8, c=64, h=w=256, K=5, pad=2. All fp32: the pipeline moves ~700MB vs
// ~3.3 GFLOP -> bandwidth-bound at 23.3 TB/s, so the wins are (a) LDS tiling
// to kill the 25x stencil read amplification, (b) TDM async tile staging
// (tensor_load_to_lds writes LDS directly, no VGPR round-trip), and (c) the
// one real GEMM (tokernel [8,64]x[64,1600]) on V_WMMA_F32_16X16X4_F32.

#define CCH   64
#define KK    5
#define PAD   2
#define USE_TDM 1

typedef __attribute__((ext_vector_type(2))) float v2f;
typedef __attribute__((ext_vector_type(8))) float v8f;
typedef __attribute__((ext_vector_type(4))) unsigned int u32x4;
typedef __attribute__((ext_vector_type(8))) int i32x8;
typedef __attribute__((ext_vector_type(4))) int i32x4;

// ---------------- avgpool 2x2 stride 2 (per plane) ----------------
__global__ void avgpool2_k(const float* __restrict__ in, float* __restrict__ out,
                           int H, int W) {
  const int ow = W >> 1, oh = H >> 1;
  const long plane = blockIdx.z;
  const int idx = blockIdx.x * blockDim.x + threadIdx.x;
  if (idx >= ow * oh) return;
  const int x = idx % ow, y = idx / ow;
  const float* p = in + plane * (long)H * W + (long)(2 * y) * W + 2 * x;
  out[plane * (long)oh * ow + idx] = 0.25f * (p[0] + p[1] + p[W] + p[W + 1]);
}

// ---------------- maxpool 2x2 stride 2 (per plane) ----------------
__global__ void maxpool2_k(const float* __restrict__ in, float* __restrict__ out,
                           int H, int W) {
  const int ow = W >> 1, oh = H >> 1;
  const long plane = blockIdx.z;
  const int idx = blockIdx.x * blockDim.x + threadIdx.x;
  if (idx >= ow * oh) return;
  const int x = idx % ow, y = idx / ow;
  const float* p = in + plane * (long)H * W + (long)(2 * y) * W + 2 * x;
  out[plane * (long)oh * ow + idx] =
      fmaxf(fmaxf(p[0], p[1]), fmaxf(p[W], p[W + 1]));
}

// ---------------- depthwise 5x5 conv, LDS-tiled, TDM-staged ----------------
// PER_BC=false: weights indexed by channel (wt + c*25), static convs.
// PER_BC=true : weights indexed by plane b*C+c (dynamic kernels).
// Block = (32,8) threads; tile 32x8 outputs, halo 2 -> 36x12 f32 in LDS.
// Interior tiles: one TDM tensor_load_to_lds (2D tile DMA, EXEC-independent),
// wave 0 issues + waits TENSORcnt, then workgroup barrier. Border tiles:
// cooperative zero-padded load.
template <bool PER_BC>
__global__ void dwconv5_k(const float* __restrict__ in, const float* __restrict__ wt,
                          const float* __restrict__ bi, float* __restrict__ out,
                          int C, int H, int W) {
  const int TW = 32, TH = 8;
  const int LW = TW + 2 * PAD;   // 36
  const int LH = TH + 2 * PAD;   // 12
  __shared__ float tile[LH][LW];
  const int bx = blockIdx.x * TW, by = blockIdx.y * TH;
  const int plane = blockIdx.z;          // b*C + c
  const int c = plane % C;
  const float* ip = in + (long)plane * H * W;
  const float* wp = wt + (PER_BC ? (long)plane * 25 : (long)c * 25);

  // Weights are uniform across the block -> scalar (SGPR) broadcast loads.
  float w[25];
#pragma unroll
  for (int i = 0; i < 25; ++i) w[i] = wp[i];
  const float bias = bi[c];

  const bool interior = (bx >= PAD) && (bx + TW + PAD <= W) &&
                        (by >= PAD) && (by + TH + PAD <= H);
#if USE_TDM
  if (interior) {
    if (threadIdx.y == 0) {   // wave 0 issues one TDM op (EXEC ignored)
      // LDS byte offset of tile[0][0]: generic -> AS(3) -> ptrtoint.
      const unsigned lds_off =
          (unsigned)(size_t)(__attribute__((address_space(3))) void*)&tile[0][0];
      const unsigned long long ga =
          (unsigned long long)(const void*)(ip + (long)(by - PAD) * W + (bx - PAD));
      // D# group 0: count=1 | lds_addr | global_addr | type=2 (ISA 8.3)
      u32x4 g0;
      g0[0] = 1u;                                   // count=1, user mode
      g0[1] = lds_off;                              // lds_addr (bytes)
      g0[2] = (unsigned)(ga & 0xFFFFFFFFull);       // global_addr[31:0]
      g0[3] = (unsigned)((ga >> 32) & 0x01FFFFFFull) | (2u << 30); // [56:32]|type=2
      // D# group 1 (ISA 8.4): data_size=4B, tensor WxH, tile 36x12, stride0=W
      i32x8 g1;
      g1[0] = (int)(2u << 16);                      // data_size=2 (4 bytes)
      g1[1] = (int)(((unsigned)W & 0xFFFFu) << 16); // tensor_dim0[15:0]
      g1[2] = (int)((((unsigned)W >> 16) & 0xFFFFu) |
                    (((unsigned)H & 0xFFFFu) << 16));// dim0[31:16] | dim1[15:0]
      g1[3] = (int)((((unsigned)H >> 16) & 0xFFFFu) |
                    ((unsigned)LW << 16));           // dim1[31:16] | tile_dim0=36
      g1[4] = (int)(unsigned)LH;                     // tile_dim1=12, tile_dim2=0
      g1[5] = (int)(unsigned)W;                      // tensor_dim0_stride[31:0]
      g1[6] = 0;                                     // stride0[47:32]|stride1[15:0]
      g1[7] = 0;                                     // stride1[47:16]
      const i32x4 z4 = {0, 0, 0, 0};                 // groups 2/3 unused (2D)
      const i32x8 z8 = {0, 0, 0, 0, 0, 0, 0, 0};     // 6-arg toolchain extra group
      __builtin_amdgcn_tensor_load_to_lds(g0, g1, z4, z4, z8, 0);
      __builtin_amdgcn_s_wait_tensorcnt(0);
    }
  } else
#endif
  {
    // Cooperative tile load with zero padding at image borders.
    const int tid = threadIdx.y * TW + threadIdx.x;
    for (int t = tid; t < LH * LW; t += TW * TH) {
      const int ly = t / LW, lx = t % LW;
      const int gy = by + ly - PAD, gx = bx + lx - PAD;
      float v = 0.0f;
      if (gy >= 0 && gy < H && gx >= 0 && gx < W) v = ip[(long)gy * W + gx];
      tile[ly][lx] = v;
    }
  }
  __syncthreads();

  const int ox = bx + threadIdx.x, oy = by + threadIdx.y;
  if (ox < W && oy < H) {
    float acc = bias;
#pragma unroll
    for (int i = 0; i < KK; ++i)
#pragma unroll
      for (int j = 0; j < KK; ++j)
        acc = fmaf(tile[threadIdx.y + i][threadIdx.x + j], w[i * 5 + j], acc);
    out[(long)plane * H * W + (long)oy * W + ox] = acc;
  }
}

// ---------------- global average pool (block per plane) ----------------
__global__ void gavg_k(const float* __restrict__ in, float* __restrict__ out, int HW) {
  __shared__ float red[256];
  const long plane = blockIdx.x;
  const float* p = in + plane * HW;
  float s = 0.0f;
  for (int i = threadIdx.x; i < HW; i += 256) s += p[i];
  red[threadIdx.x] = s;
  __syncthreads();
  for (int off = 128; off > 0; off >>= 1) {
    if ((int)threadIdx.x < off) red[threadIdx.x] += red[threadIdx.x + off];
    __syncthreads();
  }
  if (threadIdx.x == 0) out[plane] = red[0] * (1.0f / HW);
}

// ---------------- tokernel GEMM via V_WMMA_F32_16X16X4_F32 ----------------
// kern[b,o] = sum_k g[b,k] * wk[o,k] + bk[o];  b<8 (padded to M=16), o<1600.
// One wave per 16-wide N tile; 16 WMMA steps over K=64.
// A 16x4 f32 layout (ISA 7.12.2): lane M = lane&15; half selects K pair.
// Branchless: A rows 8..15 load a clamped (valid) row then select to zero.
__global__ void tokernel_wmma_k(const float* __restrict__ g,    // [8,64]
                                const float* __restrict__ wk,   // [1600,64]
                                const float* __restrict__ bk,   // [1600]
                                float* __restrict__ kern) {     // [8,1600]
  const int n0 = blockIdx.x * 16;
  const int lane = threadIdx.x;
  const int half = lane >> 4;       // 0: lanes 0-15, 1: lanes 16-31
  const int l15 = lane & 15;
  const bool live = (l15 < 8);
  const v2f* ga = (const v2f*)(g + (l15 & 7) * 64);            // clamped row
  const v2f* gb = (const v2f*)(wk + (long)(n0 + l15) * 64);    // B[k][n]=wk[n][k]
  const v2f zz = {0.0f, 0.0f};
  v8f acc = {};
#pragma unroll
  for (int kk = 0; kk < 16; ++kk) {          // K chunks of 4
    v2f a = ga[2 * kk + half];               // global_load_b64
    a = live ? a : zz;                       // v_cndmask, no branch
    const v2f b = gb[2 * kk + half];         // global_load_b64
    acc = __builtin_amdgcn_wmma_f32_16x16x4_f32(
        /*neg_a=*/false, a, /*neg_b=*/false, b,
        /*c_mod=*/(short)0, acc, /*reuse_a=*/false, /*reuse_b=*/false);
  }
  if (half == 0) {
    const float bias = bk[n0 + l15];
#pragma unroll
    for (int r = 0; r < 8; ++r)                       // rows M=0..7 == batch b
      kern[(long)r * 1600 + n0 + l15] = acc[r] + bias;
  }
}

// ---------------- driver ----------------
extern "C" void kernel_launch(void* const* d_in, const int* in_sizes, int n_in,
                              void* d_out, int out_size, void* d_ws, size_t ws_size,
                              hipStream_t stream) {
  const float* x    = (const float*)d_in[0];  // [8,64,256,256]
  const float* w1   = (const float*)d_in[1];  // [3,64,1,5,5]
  const float* b1   = (const float*)d_in[2];  // [3,64]
  const float* w2   = (const float*)d_in[3];  // [3,64,1,5,5]
  const float* b2   = (const float*)d_in[4];  // [3,64]
  const float* wk   = (const float*)d_in[5];  // [1600,64]
  const float* bk   = (const float*)d_in[6];  // [1600]
  const float* bias = (const float*)d_in[7];  // [64]
  float* out = (float*)d_out;

  const int B = 8, C = CCH;
  const int planes = B * C;                       // 512

  // workspace layout (floats), ~84 MB total
  float* ws = (float*)d_ws;
  float* t0   = ws;                               // 512*128*128
  float* t1   = t0 + (long)planes * 128 * 128;
  float* p0   = t1 + (long)planes * 128 * 128;    // 512*64*64
  float* p1   = p0 + (long)planes * 64 * 64;
  float* gbuf = p1 + (long)planes * 64 * 64;      // 512
  float* kern = gbuf + 512;                       // 8*1600

  // 1) avgpool 2x2: x[256^2] -> t0[128^2]
  avgpool2_k<<<dim3((128 * 128 + 255) / 256, 1, planes), 256, 0, stream>>>(x, t0, 256, 256);

  // 2) Block1: three 5x5 depthwise convs @128^2 (ping-pong t0<->t1)
  {
    dim3 blk(32, 8), grd(128 / 32, 128 / 8, planes);
    dwconv5_k<false><<<grd, blk, 0, stream>>>(t0, w1 + 0 * C * 25, b1 + 0 * C, t1, C, 128, 128);
    dwconv5_k<false><<<grd, blk, 0, stream>>>(t1, w1 + 1 * C * 25, b1 + 1 * C, t0, C, 128, 128);
    dwconv5_k<false><<<grd, blk, 0, stream>>>(t0, w1 + 2 * C * 25, b1 + 2 * C, t1, C, 128, 128);
  }

  // 3) maxpool 2x2: t1[128^2] -> p0[64^2]
  maxpool2_k<<<dim3((64 * 64 + 255) / 256, 1, planes), 256, 0, stream>>>(t1, p0, 128, 128);

  // 4) Block2: three 5x5 depthwise convs @64^2 (ping-pong p0<->p1)
  {
    dim3 blk(32, 8), grd(64 / 32, 64 / 8, planes);
    dwconv5_k<false><<<grd, blk, 0, stream>>>(p0, w2 + 0 * C * 25, b2 + 0 * C, p1, C, 64, 64);
    dwconv5_k<false><<<grd, blk, 0, stream>>>(p1, w2 + 1 * C * 25, b2 + 1 * C, p0, C, 64, 64);
    dwconv5_k<false><<<grd, blk, 0, stream>>>(p0, w2 + 2 * C * 25, b2 + 2 * C, p1, C, 64, 64);
  }

  // 5) global average pool: p1 -> gbuf[8,64]
  gavg_k<<<planes, 256, 0, stream>>>(p1, gbuf, 64 * 64);

  // 6) per-sample kernels via WMMA f32 GEMM: kern[8,1600] = g @ wk^T + bk
  tokernel_wmma_k<<<1600 / 16, 32, 0, stream>>>(gbuf, wk, bk, kern);

  // 7) dynamic depthwise conv on full-res x, fused bias -> out
  {
    dim3 blk(32, 8), grd(256 / 32, 256 / 8, planes);
    dwconv5_k<true><<<grd, blk, 0, stream>>>(x, kern, bias, out, C, 256, 256);
  }
}